// IntelligentPatchSelector_1649267442290
// MI455X (gfx1250) — compile-verified
//
#include <hip/hip_runtime.h>
#include <hip/hip_bf16.h>
#include <math.h>

// Problem constants (fixed by reference setup_inputs)
#define BB 8
#define NN 8192
#define DD 512
#define KMAX 512

__constant__ const float DIV_W = 0.3f;
__constant__ const float UNC_W = 0.3f;
// reciprocal of (N-1): compile-time constant, avoids v_div_scale/v_rcp chains
#define INV_NM1 (1.0f / 8191.0f)

typedef float v2f __attribute__((ext_vector_type(2)));
typedef float v8f __attribute__((ext_vector_type(8)));

// ---------------------------------------------------------------------------
// Pass 1: per-row inv-norm + per-batch column sum of normalized rows.
// One wave per row; lane loads 4x float4 (coalesced b128), wave-reduces sum
// of squares, then accumulates v*inv into a per-wave LDS partial of s.
// ---------------------------------------------------------------------------
__global__ __launch_bounds__(256) void k_norm_sum(const float* __restrict__ x,
                                                  float* __restrict__ inv_norm,
                                                  float* __restrict__ s) {
    const int b    = blockIdx.y;
    const int lane = threadIdx.x & 31;
    const int wave = threadIdx.x >> 5;          // 0..7
    __shared__ float spart[8][DD];              // 16 KB: per-wave partial of s

    for (int d = threadIdx.x; d < 8 * DD; d += blockDim.x)
        ((float*)spart)[d] = 0.0f;
    __syncthreads();

    const int rowsPerBlock = NN / gridDim.x;    // 32 blocks/batch -> 256 rows
    const int rowBase      = blockIdx.x * rowsPerBlock;
    const float* xb        = x + (size_t)b * NN * DD;

    for (int r = rowBase + wave; r < rowBase + rowsPerBlock; r += 8) {
        const float4* row = (const float4*)(xb + (size_t)r * DD);
        float4 v[4];
        float  ss = 0.0f;
#pragma unroll
        for (int c4 = 0; c4 < 4; ++c4) {
            v[c4] = row[c4 * 32 + lane];        // b128, fully coalesced
            ss += v[c4].x * v[c4].x + v[c4].y * v[c4].y +
                  v[c4].z * v[c4].z + v[c4].w * v[c4].w;
        }
#pragma unroll
        for (int m = 16; m >= 1; m >>= 1) ss += __shfl_xor(ss, m, 32);
        const float inv = 1.0f / fmaxf(sqrtf(ss), 1e-12f);
        if (lane == 0) inv_norm[(size_t)b * NN + r] = inv;
#pragma unroll
        for (int c4 = 0; c4 < 4; ++c4) {
            const int d0 = c4 * 128 + lane * 4; // each lane owns distinct dims
            spart[wave][d0 + 0] += v[c4].x * inv;
            spart[wave][d0 + 1] += v[c4].y * inv;
            spart[wave][d0 + 2] += v[c4].z * inv;
            spart[wave][d0 + 3] += v[c4].w * inv;
        }
    }
    __syncthreads();
    for (int d = threadIdx.x; d < DD; d += blockDim.x) {
        float acc = 0.0f;
#pragma unroll
        for (int w = 0; w < 8; ++w) acc += spart[w][d];
        atomicAdd(&s[b * DD + d], acc);         // 512 atomics / block
    }
}

// ---------------------------------------------------------------------------
// relevance + 0.3 * entropy(softmax) per instance (C == 2)
// ---------------------------------------------------------------------------
__global__ void k_ru(const float* __restrict__ c, float* __restrict__ ru, int total) {
    const int i = blockIdx.x * blockDim.x + threadIdx.x;
    if (i >= total) return;
    const float c0 = c[2 * i], c1 = c[2 * i + 1];
    const float rel = fmaxf(c0, c1);
    const float e0 = expf(c0 - rel), e1 = expf(c1 - rel);
    const float isum = 1.0f / (e0 + e1);
    const float p0 = e0 * isum, p1 = e1 * isum;
    const float unc = -(p0 * logf(p0 + 1e-8f) + p1 * logf(p1 + 1e-8f));
    ru[i] = rel + UNC_W * unc;
}

// ---------------------------------------------------------------------------
// Pass 2: row_sums = (x_row . s) * inv_norm via V_WMMA_F32_16X16X4_F32.
// A-frag (16x4 f32, documented layout): VGPR0 = {K0 | K2}, VGPR1 = {K1 | K3}
//   -> lane (h = lane>>4) loads float2 x[row = base + (lane&15)][k0 + 2h].
// B-frag (4x16, row-striped like C/D): VGPR r = {K=r | K=r+2}; we broadcast
//   the s chunk into all 16 columns, so b = float2 s[k0 + 2h].
// All 16 output columns are identical; lane 0 / lane 16 hold rows 0-7 / 8-15.
// ---------------------------------------------------------------------------
__global__ __launch_bounds__(256) void k_dot_combine(const float* __restrict__ x,
                                                     const float* __restrict__ s,
                                                     const float* __restrict__ inv_norm,
                                                     const float* __restrict__ ru,
                                                     float* __restrict__ combined) {
    const int b    = blockIdx.y;
    const int lane = threadIdx.x & 31;
    const int wave = threadIdx.x >> 5;
    __shared__ float sl[DD];                    // s[b] staged once per block
    for (int d = threadIdx.x; d < DD; d += blockDim.x) sl[d] = s[b * DD + d];
    __syncthreads();

    const int rowBase   = blockIdx.x * 128 + wave * 16;  // 8 waves x 16 rows
    const float* xrow   = x + (size_t)b * NN * DD
                            + (size_t)(rowBase + (lane & 15)) * DD;
    const int koff      = (lane >> 4) * 2;

    v8f acc = {};
#pragma unroll 4
    for (int k0 = 0; k0 < DD; k0 += 4) {
        v2f a  = *(const v2f*)(xrow + k0 + koff);
        v2f bb = *(const v2f*)(&sl[k0 + koff]);
        acc = __builtin_amdgcn_wmma_f32_16x16x4_f32(
            /*neg_a=*/false, a, /*neg_b=*/false, bb,
            /*c_mod=*/(short)0, acc, /*reuse_a=*/false, /*reuse_b=*/false);
    }

    if ((lane & 15) == 0) {                      // lanes 0 and 16
        const int    half = lane >> 4;
        const size_t base = (size_t)b * NN + rowBase + half * 8;
#pragma unroll
        for (int r = 0; r < 8; ++r) {
            const float rs  = acc[r] * inv_norm[base + r]; // x_hat . s
            const float avg = (rs - 1.0f) * INV_NM1;       // mul, not divide
            combined[base + r] = ru[base + r] + DIV_W * (1.0f - avg);
        }
    }
}

// ---------------------------------------------------------------------------
// Softmax over N per batch (one 1024-thread block per batch).
// Weights land after the B*K index region of d_out; K read on-device.
// ---------------------------------------------------------------------------
__global__ __launch_bounds__(1024) void k_softmax(const float* __restrict__ combined,
                                                  const int* __restrict__ lam,
                                                  float* __restrict__ out) {
    const int b   = blockIdx.x;
    const int tid = threadIdx.x;
    __shared__ float red[32];
    __shared__ float bcast;
    const float* cb = combined + (size_t)b * NN;

    float lv[8];
    float m = -3.4e38f;
#pragma unroll
    for (int k = 0; k < 8; ++k) { lv[k] = cb[tid + k * 1024]; m = fmaxf(m, lv[k]); }
#pragma unroll
    for (int s2 = 16; s2 >= 1; s2 >>= 1) m = fmaxf(m, __shfl_xor(m, s2, 32));
    if ((tid & 31) == 0) red[tid >> 5] = m;
    __syncthreads();
    if (tid < 32) {
        float v = red[tid];
#pragma unroll
        for (int s2 = 16; s2 >= 1; s2 >>= 1) v = fmaxf(v, __shfl_xor(v, s2, 32));
        if (tid == 0) bcast = v;
    }
    __syncthreads();
    m = bcast;

    float ex[8];
    float sum = 0.0f;
#pragma unroll
    for (int k = 0; k < 8; ++k) { ex[k] = expf(lv[k] - m); sum += ex[k]; }
#pragma unroll
    for (int s2 = 16; s2 >= 1; s2 >>= 1) sum += __shfl_xor(sum, s2, 32);
    if ((tid & 31) == 0) red[tid >> 5] = sum;
    __syncthreads();
    if (tid < 32) {
        float v = red[tid];
#pragma unroll
        for (int s2 = 16; s2 >= 1; s2 >>= 1) v += __shfl_xor(v, s2, 32);
        if (tid == 0) bcast = v;
    }
    __syncthreads();
    const float inv = 1.0f / bcast;

    const int K = min(lam[0], NN);
    float* w = out + (size_t)BB * K + (size_t)b * NN;
#pragma unroll
    for (int k = 0; k < 8; ++k) w[tid + k * 1024] = ex[k] * inv;
}

// ---------------------------------------------------------------------------
// Stable top-K by rank: rank(i) = #{v_j > v_i} + #{j < i : v_j == v_i}.
// Matches jax.lax.top_k ordering (descending, ties -> lower index).
// Indices emitted as float(index) into d_out[0 .. B*K).
// ---------------------------------------------------------------------------
__global__ __launch_bounds__(256) void k_topk(const float* __restrict__ combined,
                                              const int* __restrict__ lam,
                                              float* __restrict__ out) {
    const int b = blockIdx.y;
    __shared__ float vals[NN];                  // 32 KB
    const float* cb = combined + (size_t)b * NN;
    for (int i = threadIdx.x; i < NN; i += 256) vals[i] = cb[i];
    __syncthreads();

    const int   i = blockIdx.x * 256 + threadIdx.x;
    const float v = vals[i];
    int gt = 0, eqb = 0;
#pragma unroll 8
    for (int j = 0; j < NN; j += 4) {
        const float4 q = *(const float4*)&vals[j];   // LDS broadcast reads
        gt  += (int)(q.x > v) + (int)(q.y > v) + (int)(q.z > v) + (int)(q.w > v);
        eqb += (int)((q.x == v) & (j + 0 < i)) + (int)((q.y == v) & (j + 1 < i))
             + (int)((q.z == v) & (j + 2 < i)) + (int)((q.w == v) & (j + 3 < i));
    }
    const int rank = gt + eqb;
    const int K = min(lam[0], NN);
    if (rank < K) out[(size_t)b * K + rank] = (float)i;
}

// ---------------------------------------------------------------------------
extern "C" void kernel_launch(void* const* d_in, const int* in_sizes, int n_in,
                              void* d_out, int out_size, void* d_ws, size_t ws_size,
                              hipStream_t stream) {
    const float* x   = (const float*)d_in[0];   // [B, N, D] f32
    const float* c   = (const float*)d_in[1];   // [B, N, 2] f32
    const int*   lam = (const int*)d_in[2];     // scalar big_lambda
    float* out = (float*)d_out;
    float* ws  = (float*)d_ws;

    // workspace layout (floats): inv_norm[B*N] | s[B*D] | ru[B*N] | combined[B*N]
    float* inv_norm = ws;
    float* s        = inv_norm + (size_t)BB * NN;
    float* ru       = s + (size_t)BB * DD;
    float* combined = ru + (size_t)BB * NN;

    (void)in_sizes; (void)n_in; (void)out_size; (void)ws_size;

    hipMemsetAsync(s, 0, (size_t)BB * DD * sizeof(float), stream);
    k_norm_sum   <<<dim3(32, BB), 256, 0, stream>>>(x, inv_norm, s);
    k_ru         <<<(BB * NN + 255) / 256, 256, 0, stream>>>(c, ru, BB * NN);
    k_dot_combine<<<dim3(NN / 128, BB), 256, 0, stream>>>(x, s, inv_norm, ru, combined);
    k_softmax    <<<BB, 1024, 0, stream>>>(combined, lam, out);
    k_topk       <<<dim3(NN / 256, BB), 256, 0, stream>>>(combined, lam, out);
}